// Set2Set_49478023250671
// MI455X (gfx1250) — compile-verified
//
#include <hip/hip_runtime.h>
#include <math.h>

// Problem constants (match reference)
#define DD   128        // in_channels
#define BB   16384      // number of graphs
#define NN   500000     // nodes
#define GATE 512        // 4*D
#define QS   256        // 2*D
#define STEPS 3
#define EPS_SM 1e-16f

typedef __attribute__((ext_vector_type(2))) float v2f;
typedef __attribute__((ext_vector_type(8))) float v8f;

__device__ __forceinline__ v8f wmma_f32(v2f a, v2f b, v8f c) {
    // D = A(16x4) x B(4x16) + C(16x16), fp32, wave32
    return __builtin_amdgcn_wmma_f32_16x16x4_f32(
        /*neg_a=*/false, a, /*neg_b=*/false, b,
        /*c_mod=*/(short)0, c, /*reuse_a=*/false, /*reuse_b=*/false);
}

// ---------------------------------------------------------------------------
// 0) Zero h, c (ws) and q_star (d_out)
// ---------------------------------------------------------------------------
__global__ __launch_bounds__(256) void init_zero(float* __restrict__ hc,
                                                 float* __restrict__ qstar) {
    int idx = blockIdx.x * 256 + threadIdx.x;
    const int HC = 2 * BB * DD;         // h then c
    if (idx < HC) hc[idx] = 0.0f;
    if (idx < BB * QS) qstar[idx] = 0.0f;
}

// ---------------------------------------------------------------------------
// 1) Segment ranges: batch is sorted; binary-search [start,end) per segment b
// ---------------------------------------------------------------------------
__device__ __forceinline__ int lower_bound_i32(const int* __restrict__ a, int n, int key) {
    int lo = 0, hi = n;
    while (lo < hi) {
        int mid = (lo + hi) >> 1;
        if (a[mid] < key) lo = mid + 1; else hi = mid;
    }
    return lo;
}

__global__ __launch_bounds__(256) void seg_ranges(const int* __restrict__ batch,
                                                  int2* __restrict__ ranges) {
    int b = blockIdx.x * 256 + threadIdx.x;
    if (b >= BB) return;
    int s0 = lower_bound_i32(batch, NN, b);
    int s1 = lower_bound_i32(batch, NN, b + 1);
    ranges[b] = make_int2(s0, s1);
}

// ---------------------------------------------------------------------------
// 2) LSTM gate GEMM via V_WMMA_F32_16X16X4_F32
//    gates[B,512] = q_star[B,256] @ W_ih^T + h[B,128] @ W_hh^T + (b_ih + b_hh)
//    Per wave: 64(M) x 32(N) tile -> 4x2 accumulators, K = 256 then 128.
//    K unrolled x4 so byte offsets fold into the 24-bit VMEM immediate.
// ---------------------------------------------------------------------------
__global__ __launch_bounds__(128) void lstm_gates_wmma(
    const float* __restrict__ qstar,   // [B, 256]
    const float* __restrict__ h,       // [B, 128]
    const float* __restrict__ W_ih,    // [512, 256]
    const float* __restrict__ W_hh,    // [512, 128]
    const float* __restrict__ b_ih,    // [512]
    const float* __restrict__ b_hh,    // [512]
    float* __restrict__ gates)         // [B, 512]
{
    const int lane = threadIdx.x & 31;
    const int wave = threadIdx.x >> 5;
    const int tile = blockIdx.x * 4 + wave;          // 4096 wave-tiles total
    const int N_GRP = GATE / 32;                     // 16
    const int mBase = (tile / N_GRP) * 64;
    const int nBase = (tile % N_GRP) * 32;
    const int hl = lane & 15;                        // M (A) / N (B) index
    const int kh = (lane >> 4) << 1;                 // K sub-offset: 0 or 2
    const int hi8 = (lane >> 4) << 3;                // C-row offset: 0 or 8

    v8f acc[4][2];
    v8f z = {};
    #pragma unroll
    for (int mi = 0; mi < 4; ++mi)
        #pragma unroll
        for (int nj = 0; nj < 2; ++nj) acc[mi][nj] = z;

    // Per-lane base pointers (advance by constant; k*4B folds into IOFFSET)
    const float* aP[4];
    const float* bP[2];
    #pragma unroll
    for (int mi = 0; mi < 4; ++mi)
        aP[mi] = qstar + (size_t)(mBase + mi * 16 + hl) * 256 + kh;
    #pragma unroll
    for (int nj = 0; nj < 2; ++nj)
        bP[nj] = W_ih + (size_t)(nBase + nj * 16 + hl) * 256 + kh;

    // Phase 1: K over q_star (ld 256) with W_ih
    #pragma unroll 4
    for (int k = 0; k < 256; k += 4) {
        v2f a[4], bf[2];
        #pragma unroll
        for (int mi = 0; mi < 4; ++mi) a[mi] = *(const v2f*)(aP[mi] + k);
        #pragma unroll
        for (int nj = 0; nj < 2; ++nj) bf[nj] = *(const v2f*)(bP[nj] + k);
        #pragma unroll
        for (int mi = 0; mi < 4; ++mi)
            #pragma unroll
            for (int nj = 0; nj < 2; ++nj)
                acc[mi][nj] = wmma_f32(a[mi], bf[nj], acc[mi][nj]);
    }

    // Phase 2: K over h (ld 128) with W_hh
    #pragma unroll
    for (int mi = 0; mi < 4; ++mi)
        aP[mi] = h + (size_t)(mBase + mi * 16 + hl) * 128 + kh;
    #pragma unroll
    for (int nj = 0; nj < 2; ++nj)
        bP[nj] = W_hh + (size_t)(nBase + nj * 16 + hl) * 128 + kh;

    #pragma unroll 4
    for (int k = 0; k < 128; k += 4) {
        v2f a[4], bf[2];
        #pragma unroll
        for (int mi = 0; mi < 4; ++mi) a[mi] = *(const v2f*)(aP[mi] + k);
        #pragma unroll
        for (int nj = 0; nj < 2; ++nj) bf[nj] = *(const v2f*)(bP[nj] + k);
        #pragma unroll
        for (int mi = 0; mi < 4; ++mi)
            #pragma unroll
            for (int nj = 0; nj < 2; ++nj)
                acc[mi][nj] = wmma_f32(a[mi], bf[nj], acc[mi][nj]);
    }

    // Store + bias. C/D layout: VGPR i -> row (i + hi8), col = hl within tile.
    #pragma unroll
    for (int nj = 0; nj < 2; ++nj) {
        const int nCol = nBase + nj * 16 + hl;
        const float bias = b_ih[nCol] + b_hh[nCol];
        #pragma unroll
        for (int mi = 0; mi < 4; ++mi) {
            const int mRow = mBase + mi * 16 + hi8;
            #pragma unroll
            for (int i = 0; i < 8; ++i)
                gates[(size_t)(mRow + i) * GATE + nCol] = acc[mi][nj][i] + bias;
        }
    }
}

// ---------------------------------------------------------------------------
// 3) LSTM elementwise: c' = sig(f)*c + sig(i)*tanh(g); h' = sig(o)*tanh(c')
//    Also writes q_star[:, 0:128] = h'
// ---------------------------------------------------------------------------
__device__ __forceinline__ float sigm(float x) { return 1.0f / (1.0f + expf(-x)); }

__global__ __launch_bounds__(256) void lstm_activation(
    const float* __restrict__ gates,  // [B, 512]
    float* __restrict__ h,            // [B, 128]
    float* __restrict__ c,            // [B, 128]
    float* __restrict__ qstar)        // [B, 256]
{
    int idx = blockIdx.x * 256 + threadIdx.x;   // B*D threads
    if (idx >= BB * DD) return;
    int b = idx >> 7;          // /128
    int d = idx & 127;
    const float* g = gates + (size_t)b * GATE;
    float gi = g[d];
    float gf = g[d + 128];
    float gg = g[d + 256];
    float go = g[d + 384];
    float cn = sigm(gf) * c[idx] + sigm(gi) * tanhf(gg);
    float hn = sigm(go) * tanhf(cn);
    c[idx] = cn;
    h[idx] = hn;
    qstar[(size_t)b * QS + d] = hn;
}

// ---------------------------------------------------------------------------
// 4) Fused per-segment attention: one wave per segment (batch sorted).
//    Pass 1: running max of e[n] = <x[n], h[b]>. Pass 2 (x hot in cache):
//    s = sum exp(e-m); r = sum exp(e-m)*x;  q_star[b,128:256] = r/(s+eps)
// ---------------------------------------------------------------------------
__global__ __launch_bounds__(256) void seg_attention(
    const float* __restrict__ x,       // [N, 128]
    const float* __restrict__ h,       // [B, 128]
    const int2* __restrict__ ranges,   // [B]
    float* __restrict__ qstar)         // [B, 256]
{
    const int lane = threadIdx.x & 31;
    const int wave = threadIdx.x >> 5;
    const int b = blockIdx.x * 8 + wave;
    const int2 r = ranges[b];

    const float* hrow = h + (size_t)b * DD + lane * 4;
    float h0 = hrow[0], h1 = hrow[1], h2 = hrow[2], h3 = hrow[3];

    // Pass 1: segment max (empty segment -> m = -inf, loops skipped, r = 0)
    float m = -INFINITY;
    for (int n = r.x; n < r.y; ++n) {
        const float* xr = x + (size_t)n * DD + lane * 4;
        if (n + 2 < r.y)
            __builtin_prefetch(x + (size_t)(n + 2) * DD + lane * 4, 0, 1);
        float p = xr[0] * h0 + xr[1] * h1 + xr[2] * h2 + xr[3] * h3;
        p += __shfl_xor(p, 16, 32);
        p += __shfl_xor(p, 8, 32);
        p += __shfl_xor(p, 4, 32);
        p += __shfl_xor(p, 2, 32);
        p += __shfl_xor(p, 1, 32);
        m = fmaxf(m, p);
    }

    // Pass 2: weighted sums (x rows just touched -> WGP$/L2 hits)
    float s = 0.0f;
    float r0 = 0.0f, r1 = 0.0f, r2 = 0.0f, r3 = 0.0f;
    for (int n = r.x; n < r.y; ++n) {
        const float* xr = x + (size_t)n * DD + lane * 4;
        float x0 = xr[0], x1 = xr[1], x2 = xr[2], x3 = xr[3];
        float p = x0 * h0 + x1 * h1 + x2 * h2 + x3 * h3;
        p += __shfl_xor(p, 16, 32);
        p += __shfl_xor(p, 8, 32);
        p += __shfl_xor(p, 4, 32);
        p += __shfl_xor(p, 2, 32);
        p += __shfl_xor(p, 1, 32);
        float w = expf(p - m);
        s += w;
        r0 += w * x0; r1 += w * x1; r2 += w * x2; r3 += w * x3;
    }

    float inv = 1.0f / (s + EPS_SM);
    float* out = qstar + (size_t)b * QS + DD + lane * 4;
    out[0] = r0 * inv;
    out[1] = r1 * inv;
    out[2] = r2 * inv;
    out[3] = r3 * inv;
}

// ---------------------------------------------------------------------------
// Host-side orchestration
// ---------------------------------------------------------------------------
extern "C" void kernel_launch(void* const* d_in, const int* in_sizes, int n_in,
                              void* d_out, int out_size, void* d_ws, size_t ws_size,
                              hipStream_t stream) {
    const float* x     = (const float*)d_in[0];   // [N, 128]
    const int*   batch = (const int*)  d_in[1];   // [N]
    const float* W_ih  = (const float*)d_in[2];   // [512, 256]
    const float* W_hh  = (const float*)d_in[3];   // [512, 128]
    const float* b_ih  = (const float*)d_in[4];   // [512]
    const float* b_hh  = (const float*)d_in[5];   // [512]
    float* qstar = (float*)d_out;                 // [B, 256] = output & state

    // Workspace layout (floats)
    float* h     = (float*)d_ws;                       // B*D
    float* c     = h + (size_t)BB * DD;                // B*D
    float* gates = c + (size_t)BB * DD;                // B*4D
    int2*  ranges = (int2*)(gates + (size_t)BB * GATE);

    // Init h, c, q_star to zero (2*B*D == B*2D == 4194304 elements each)
    {
        int total = 2 * BB * DD;
        init_zero<<<(total + 255) / 256, 256, 0, stream>>>(h, qstar);
    }
    // Segment ranges (batch static across steps)
    seg_ranges<<<(BB + 255) / 256, 256, 0, stream>>>(batch, ranges);

    const int gemm_blocks = (BB / 64) * (GATE / 32) / 4;      // 1024
    const int act_blocks  = (BB * DD + 255) / 256;            // 8192
    const int attn_blocks = BB / 8;                           // 2048

    for (int step = 0; step < STEPS; ++step) {
        lstm_gates_wmma<<<gemm_blocks, 128, 0, stream>>>(
            qstar, h, W_ih, W_hh, b_ih, b_hh, gates);
        lstm_activation<<<act_blocks, 256, 0, stream>>>(gates, h, c, qstar);
        seg_attention<<<attn_blocks, 256, 0, stream>>>(x, h, ranges, qstar);
    }
}